// Modified_shift_gcn_23261542875780
// MI455X (gfx1250) — compile-verified
//
#include <hip/hip_runtime.h>

typedef __attribute__((ext_vector_type(2))) float v2f;
typedef __attribute__((ext_vector_type(8))) float v8f;

#define NB    128      // batch N
#define CIN   128
#define DOUTC 128
#define TT    512
#define VVN   4
#define NTROW (NB * TT)        // 65536 rows of h
#define AST   132              // padded LDS stride for A tile (floats)
#define YST   520              // padded LDS stride for Y tile (floats)

// ---------------------------------------------------------------------------
// k0: zero the 1024-float stats accumulators in workspace (every launch).
// ---------------------------------------------------------------------------
__global__ __launch_bounds__(1024) void zero_stats_kernel(float* g) {
  int i = threadIdx.x;
  if (i < 1024) g[i] = 0.0f;
}

// ---------------------------------------------------------------------------
// Fused gather+mask+GEMM kernel. Each block: one n, 16 t values, all 4 v
// => 64 GEMM rows x 128 output cols, K = 128 via v_wmma_f32_16x16x4_f32.
//   FINAL=false : accumulate per-Y-column sum / sumsq into gsum/gsq.
//   FINAL=true  : recompute Y, apply shift_out + BN scale/shift + residual
//                 + ReLU, store to out[N,D,T,V].
// ---------------------------------------------------------------------------
template <bool FINAL>
__global__ __launch_bounds__(256) void gcn_gemm_kernel(
    const float* __restrict__ x,      // [N, C, T, V]
    const float* __restrict__ W,      // [C, D] row-major
    const float* __restrict__ Lb,     // [D]
    const float* __restrict__ FM,     // [V*C]
    const int*   __restrict__ sin_,   // [512]
    const int*   __restrict__ sout_,  // [512]
    float* __restrict__ gsum,         // [512]
    float* __restrict__ gsq,          // [512]
    const float* __restrict__ scl,    // [512] (FINAL)
    const float* __restrict__ sft,    // [512] (FINAL)
    float* __restrict__ out)          // [N, D, T, V]
{
  // union buffer: A tile 64 x AST (8448 f) in GEMM phase, Y tile 16 x YST
  // (8320 f) in the FINAL epilogue.
  __shared__ __align__(16) float uni[64 * AST];
  __shared__ float s_mask[512];
  __shared__ int   s_sidx[512];
  __shared__ float sumBuf[512];
  __shared__ float sqBuf[512];
  __shared__ int   s_sout[512];
  __shared__ float s_scale[512];
  __shared__ float s_shift[512];

  const int tid  = threadIdx.x;
  const int lane = tid & 31;          // wave32
  const int wid  = tid >> 5;          // 8 waves
  const int blk  = blockIdx.x;        // 4096 blocks
  const int n    = blk >> 5;          // 32 blocks per n (each covers 16 t)
  const int t0   = (blk & 31) << 4;

  // ---- stage per-column tables ----
  for (int j = tid; j < 512; j += 256) {
    s_sidx[j] = sin_[j];
    s_mask[j] = tanhf(FM[j]) + 1.0f;
    if (!FINAL) { sumBuf[j] = 0.0f; sqBuf[j] = 0.0f; }
    if (FINAL)  { s_sout[j] = sout_[j]; s_scale[j] = scl[j]; s_shift[j] = sft[j]; }
  }
  __syncthreads();

  // ---- load A tile (gather via shift_in, apply mask) ----
  // GEMM row m_local = t_l*4 + v ; column c. 8192 elements, 32 per thread.
  // Lanes sweep t (stride 16B in x) for global-load friendliness.
  #pragma unroll
  for (int it = 0; it < 32; ++it) {
    int e    = it * 256 + tid;        // e = vc*16 + t_l
    int t_l  = e & 15;
    int vc   = e >> 4;                // v*128 + c
    int sidx = s_sidx[vc];
    int v2   = sidx >> 7;
    int c2   = sidx & 127;
    float val = x[(((size_t)n * CIN + c2) * TT + (t0 + t_l)) * VVN + v2] * s_mask[vc];
    int v = vc >> 7;
    int c = vc & 127;
    uni[(t_l * 4 + v) * AST + c] = val;
  }
  __syncthreads();

  // ---- WMMA GEMM: wave owns cols [wid*16, wid*16+16), all 4 row groups ----
  const int mrow = lane & 15;
  const int koff = (lane >> 4) << 1;         // 0 or 2 (A/B fragment K offset)
  const int dcol = wid * 16 + (lane & 15);

  v8f acc[4];
  #pragma unroll
  for (int rg = 0; rg < 4; ++rg)
    #pragma unroll
    for (int i = 0; i < 8; ++i) acc[rg][i] = 0.0f;

  for (int k0 = 0; k0 < CIN; k0 += 4) {
    const int kk = k0 + koff;                // even -> 8B aligned LDS reads
    v2f b;
    b.x = W[kk * DOUTC + dcol];
    b.y = W[(kk + 1) * DOUTC + dcol];
    #pragma unroll
    for (int rg = 0; rg < 4; ++rg) {
      v2f a = *(const v2f*)&uni[(rg * 16 + mrow) * AST + kk];
      acc[rg] = __builtin_amdgcn_wmma_f32_16x16x4_f32(
          false, a, false, b, (short)0, acc[rg], false, false);
    }
  }

  // ---- add bias (all 8 acc elements of a lane share column dcol) ----
  const float bias_d = Lb[dcol];
  #pragma unroll
  for (int rg = 0; rg < 4; ++rg)
    #pragma unroll
    for (int i = 0; i < 8; ++i) acc[rg][i] += bias_d;

  if (!FINAL) {
    // acc element i <-> row (i + 16*(lane>=16 ? .. ) ... v = i&3); pairs
    // (i, i+4) share Y column ycol = (i&3)*128 + dcol -> combine first.
    #pragma unroll
    for (int rg = 0; rg < 4; ++rg)
      #pragma unroll
      for (int vv = 0; vv < 4; ++vv) {
        float a0 = acc[rg][vv], a1 = acc[rg][vv + 4];
        atomicAdd(&sumBuf[vv * 128 + dcol], a0 + a1);
        atomicAdd(&sqBuf [vv * 128 + dcol], a0 * a0 + a1 * a1);
      }
    __syncthreads();
    for (int j = tid; j < 512; j += 256) {
      atomicAdd(&gsum[j], sumBuf[j]);
      atomicAdd(&gsq [j], sqBuf[j]);
    }
  } else {
    __syncthreads();   // all A-tile reads done before aliasing uni as Y tile

    // scatter acc into Y tile: Y_lds[nt_local][v*128 + d]
    const int hi = (lane >> 4) << 3;         // row +8 for lanes 16..31
    #pragma unroll
    for (int rg = 0; rg < 4; ++rg)
      #pragma unroll
      for (int i = 0; i < 8; ++i) {
        int m    = rg * 16 + i + hi;         // local GEMM row
        int ycol = (i & 3) * 128 + dcol;     // v = m&3 == i&3
        uni[(m >> 2) * YST + ycol] = acc[rg][i];
      }
    __syncthreads();

    // fused shift_out gather + BN + residual + ReLU + [N,D,T,V] store.
    // e = d*64 + t_l*4 + v : each wave writes 128B-contiguous runs.
    #pragma unroll
    for (int it = 0; it < 32; ++it) {
      int e   = it * 256 + tid;
      int d   = e >> 6;
      int rem = e & 63;
      int t_l = rem >> 2;
      int v   = rem & 3;
      int j   = v * 128 + d;
      float val = uni[t_l * YST + s_sout[j]] * s_scale[j] + s_shift[j];
      size_t addr = (((size_t)n * DOUTC + d) * TT + (t0 + t_l)) * VVN + v;
      out[addr] = fmaxf(val + x[addr], 0.0f);
    }
  }
}

// ---------------------------------------------------------------------------
// k2: fold batch stats + gamma/beta into per-output-channel scale/shift.
// ---------------------------------------------------------------------------
__global__ __launch_bounds__(512) void finalize_stats_kernel(
    const float* __restrict__ gsum, const float* __restrict__ gsq,
    const float* __restrict__ gamma, const float* __restrict__ beta,
    const int* __restrict__ sout_, float* __restrict__ scl, float* __restrict__ sft)
{
  int j = threadIdx.x;
  if (j < 512) {
    int   s    = sout_[j];
    float inv  = 1.0f / (float)NTROW;
    float mean = gsum[s] * inv;
    float var  = gsq[s] * inv - mean * mean;
    float sc   = gamma[j] * rsqrtf(var + 1e-5f);
    scl[j] = sc;
    sft[j] = beta[j] - mean * sc;
  }
}

extern "C" void kernel_launch(void* const* d_in, const int* in_sizes, int n_in,
                              void* d_out, int out_size, void* d_ws, size_t ws_size,
                              hipStream_t stream) {
  const float* x     = (const float*)d_in[0];
  const float* W     = (const float*)d_in[1];
  const float* Lb    = (const float*)d_in[2];
  const float* FM    = (const float*)d_in[3];
  const float* gamma = (const float*)d_in[4];
  const float* beta  = (const float*)d_in[5];
  const int*   sin_  = (const int*)d_in[6];
  const int*   sout_ = (const int*)d_in[7];
  float* out = (float*)d_out;

  float* ws   = (float*)d_ws;     // 8 KB used
  float* gsum = ws;               // [512]
  float* gsq  = ws + 512;         // [512]
  float* scl  = ws + 1024;        // [512]
  float* sft  = ws + 1536;        // [512]

  zero_stats_kernel<<<1, 1024, 0, stream>>>(ws);
  gcn_gemm_kernel<false><<<4096, 256, 0, stream>>>(
      x, W, Lb, FM, sin_, sout_, gsum, gsq, scl, sft, out);
  finalize_stats_kernel<<<1, 512, 0, stream>>>(gsum, gsq, gamma, beta, sout_, scl, sft);
  gcn_gemm_kernel<true><<<4096, 256, 0, stream>>>(
      x, W, Lb, FM, sin_, sout_, gsum, gsq, scl, sft, out);
}